// CrossViM_53412213293200
// MI455X (gfx1250) — compile-verified
//
#include <hip/hip_runtime.h>
#include <hip/hip_bf16.h>
#include <cmath>

// ---------------- model dims ----------------
#define L_TOK 197
#define NPATCH 196
#define DM 192
#define DI 384
#define DSN 16
#define DTRK 12
#define NDEPTH 24
#define NB 4

typedef float v2f __attribute__((ext_vector_type(2)));
typedef float v8f __attribute__((ext_vector_type(8)));

// =====================================================================
// Generic fp32 WMMA GEMM:  C = alpha * (A [+A2]) x op(B)  [+bias] [+Cadd]
//   TRANSB=0 -> B stored [N][K] (NT, i.e. "@ W.T"); TRANSB=1 -> B [K][N]
//   mode: 0 = none, 1 = softplus (runtime, epilogue only)
// One wave computes one 16x16 C tile via V_WMMA_F32_16X16X4_F32.
// Out-of-range lanes load a CLAMPED row/col: A row m only feeds D row m and
// B col n only feeds D col n, and those are masked at the C write, so the
// K-loop is completely branch-free (pure b64 loads + wmma).
// =====================================================================
template <int TRANSB, int HASA2>
__global__ __launch_bounds__(128)
void gemm16_f32(const float* __restrict__ A, int sA, int lda,
                const float* __restrict__ A2, int sA2,
                const float* __restrict__ B, int sB, int ldb,
                const float* __restrict__ bias, int sBias,
                const float* __restrict__ Cadd, int sCadd, int ldAdd,
                float* __restrict__ C, int sC, int ldc,
                int M, int N, int K, float alpha, int mode)
{
    const int z = blockIdx.z;
    A += (size_t)z * sA;
    B += (size_t)z * sB;
    if (bias) bias += (size_t)z * sBias;
    if (Cadd) Cadd += (size_t)z * sCadd;
    C += (size_t)z * sC;

    const int ntx = (N + 15) >> 4;
    const int nty = (M + 15) >> 4;
    const int wave = (blockIdx.x * (blockDim.x >> 5)) + (threadIdx.x >> 5);
    if (wave >= ntx * nty) return;
    const int tn = wave % ntx, tm = wave / ntx;
    const int lane = threadIdx.x & 31;
    const int half = lane >> 4;        // selects K pair {0,1} vs {2,3}
    const int l16  = lane & 15;
    const int ma = tm * 16 + l16;      // A row this lane feeds
    const int nb = tn * 16 + l16;      // B column this lane feeds
    const int mac = (ma < M) ? ma : (M - 1);   // clamped (garbage rows masked at store)
    const int nbc = (nb < N) ? nb : (N - 1);

    const float* Ap  = A + (size_t)mac * lda + 2 * half;
    const float* A2p = nullptr;
    if (HASA2) A2p = A2 + (size_t)z * sA2 + (size_t)mac * lda + 2 * half;
    const float* Bp = TRANSB ? (B + (size_t)(2 * half) * ldb + nbc)
                             : (B + (size_t)nbc * ldb + 2 * half);

    v8f acc = {0.f,0.f,0.f,0.f,0.f,0.f,0.f,0.f};
    #pragma unroll 4
    for (int k0 = 0; k0 < K; k0 += 4) {
        v2f a = *(const v2f*)(Ap + k0);
        if (HASA2) {
            v2f a2 = *(const v2f*)(A2p + k0);
            a.x += a2.x; a.y += a2.y;
        }
        v2f b;
        if (TRANSB) {
            b.x = Bp[(size_t)k0 * ldb];
            b.y = Bp[(size_t)(k0 + 1) * ldb];
        } else {
            b = *(const v2f*)(Bp + k0);
        }
        acc = __builtin_amdgcn_wmma_f32_16x16x4_f32(
                  false, a, false, b, (short)0, acc, false, false);
    }

    const float bval = (bias && nb < N) ? bias[nb] : 0.f;
    #pragma unroll
    for (int r = 0; r < 8; ++r) {
        const int m = tm * 16 + r + 8 * half;
        const int n = tn * 16 + l16;
        if (m < M && n < N) {
            float v = alpha * acc[r] + bval;
            if (Cadd) v += Cadd[(size_t)m * ldAdd + n];
            if (mode == 1) v = (v > 20.f) ? v : log1pf(expf(v));
            C[(size_t)m * ldc + n] = v;
        }
    }
}

// ---------------- RoPE tables (module constants in the reference) ---------
__global__ void rope_tab_k(float* __restrict__ rc, float* __restrict__ rs)
{
    int idx = blockIdx.x * blockDim.x + threadIdx.x;
    if (idx >= NPATCH * DM) return;
    int p = idx / DM, d = idx % DM;
    int py = p / 14, px = p % 14;
    int i = (d < 96) ? d : d - 96;
    float t = (float)((d < 96) ? py : px);
    float expo = (float)(2 * (i / 2)) / 96.f;
    float ang = t * powf(10000.f, -expo);
    rc[idx] = cosf(ang);
    rs[idx] = sinf(ang);
}

// ---------------- im2col (16x16 patches, stride 16) -----------------------
__global__ void im2col_k(const float* __restrict__ x1, const float* __restrict__ x2,
                         float* __restrict__ col)
{
    int idx = blockIdx.x * blockDim.x + threadIdx.x;
    const int TOT = 2 * NB * NPATCH * 768;
    if (idx >= TOT) return;
    int k = idx % 768; int t = idx / 768;
    int p = t % NPATCH; t /= NPATCH;
    int b = t % NB; int e = t / NB;
    const float* img = (e ? x2 : x1) + (size_t)b * 3 * 224 * 224;
    int ci = k / 256, rem = k % 256, iy = rem / 16, ix = rem % 16;
    int py = p / 14, px = p % 14;
    col[idx] = img[((size_t)ci * 224 + py * 16 + iy) * 224 + px * 16 + ix];
}

// ---------------- init: res = 0, h[cls row] = cls + pos[0] ----------------
__global__ void init_h_k(const float* __restrict__ cls, const float* __restrict__ pos,
                         float* __restrict__ h, float* __restrict__ res)
{
    int idx = blockIdx.x * blockDim.x + threadIdx.x;
    const int TOT = 2 * NB * L_TOK * DM;
    if (idx >= TOT) return;
    res[idx] = 0.f;
    int d = idx % DM; int t = idx / DM;
    int l = t % L_TOK; t /= L_TOK;
    int e = t / NB;
    if (l == 0) h[idx] = cls[e * DM + d] + pos[(size_t)e * L_TOK * DM + d];
}

// ---------------- fused RoPE + residual + RMSNorm (wave per token) --------
__global__ __launch_bounds__(256)
void rope_norm_k(const float* __restrict__ h, float* __restrict__ res,
                 float* __restrict__ hs, const float* __restrict__ rc,
                 const float* __restrict__ rs, const float* __restrict__ nw, int layer)
{
    const int NT = 2 * NB * L_TOK;
    int wv = (blockIdx.x * (blockDim.x >> 5)) + (threadIdx.x >> 5);
    if (wv >= NT) return;
    int e = wv / (NB * L_TOK);
    int l = wv % L_TOK;
    int lane = threadIdx.x & 31;
    size_t base = (size_t)wv * DM;
    float r[6]; float ss = 0.f;
    #pragma unroll
    for (int j = 0; j < 6; ++j) {
        int d = lane + 32 * j;
        float hv = h[base + d];
        if (l > 0) {
            int p = l - 1;
            float c = rc[p * DM + d], s = rs[p * DM + d];
            float partner = h[base + (d ^ 1)];
            float rot = (d & 1) ? partner : -partner;
            hv = hv * c + rot * s;
        }
        float rr = res[base + d] + hv;
        res[base + d] = rr;
        r[j] = rr; ss += rr * rr;
    }
    for (int o = 16; o > 0; o >>= 1) ss += __shfl_xor(ss, o, 32);
    float inv = rsqrtf(ss / (float)DM + 1e-5f);
    const float* w = nw + ((size_t)e * NDEPTH + layer) * DM;
    #pragma unroll
    for (int j = 0; j < 6; ++j) {
        int d = lane + 32 * j;
        hs[base + d] = r[j] * inv * w[d];
    }
}

// ---------------- causal depthwise conv (window 4) + SiLU, both dirs ------
__global__ void conv_silu_k(const float* __restrict__ xz, const float* __restrict__ cw,
                            const float* __restrict__ cb, float* __restrict__ xc, int layer)
{
    int idx = blockIdx.x * blockDim.x + threadIdx.x;
    const int TOT = 2 * 2 * NB * L_TOK * DI;
    if (idx >= TOT) return;
    int c = idx % DI; int t = idx / DI;
    int l = t % L_TOK; t /= L_TOK;
    int b = t % NB; t /= NB;
    int e = t % 2; int dir = t / 2;
    const float* xzp = xz + ((size_t)e * NB + b) * L_TOK * (2 * DI);
    size_t wb = ((size_t)(e * NDEPTH + layer) * 2 + dir) * DI + c;
    const float* wp = cw + wb * 4;
    float acc = cb[wb];
    #pragma unroll
    for (int k = 0; k < 4; ++k) {
        int li = l - 3 + k;
        if (li >= 0) {
            int lsrc = dir ? (L_TOK - 1 - li) : li;
            acc += xzp[(size_t)lsrc * (2 * DI) + c] * wp[k];
        }
    }
    xc[idx] = acc / (1.f + __expf(-acc));   // silu
}

// ---------------- selective scan: block = (dir,e,b), thread = channel -----
__global__ __launch_bounds__(384)
void scan_k(const float* __restrict__ xc, const float* __restrict__ dbl,
            const float* __restrict__ delta, const float* __restrict__ xz,
            const float* __restrict__ alog, const float* __restrict__ dpw,
            float* __restrict__ y, int layer)
{
    __shared__ float sh[32];                 // B(16) + C(16) per step
    int blk = blockIdx.x;                    // 0..15
    int b = blk % NB; int t = blk / NB;
    int e = t % 2; int dir = t / 2;
    int c = threadIdx.x;                     // 0..383
    size_t pbase = ((size_t)(dir * 2 + e) * NB + b) * L_TOK;
    const float* xcP  = xc    + pbase * DI;
    const float* dblP = dbl   + pbase * 48;
    const float* dlP  = delta + pbase * DI;
    float*       yP   = y     + pbase * DI;
    const float* zP   = xz + ((size_t)e * NB + b) * L_TOK * (2 * DI) + DI;
    size_t wb = ((size_t)(e * NDEPTH + layer) * 2 + dir) * DI + c;
    float Av[DSN];
    #pragma unroll
    for (int n = 0; n < DSN; ++n) Av[n] = -__expf(alog[wb * DSN + n]);
    const float Dv = dpw[wb];
    float hst[DSN];
    #pragma unroll
    for (int n = 0; n < DSN; ++n) hst[n] = 0.f;

    for (int l = 0; l < L_TOK; ++l) {
        if (threadIdx.x < 32) sh[threadIdx.x] = dblP[(size_t)l * 48 + DTRK + threadIdx.x];
        __syncthreads();
        float dlt = dlP[(size_t)l * DI + c];
        float xv  = xcP[(size_t)l * DI + c];
        float dx  = dlt * xv;
        float ys  = 0.f;
        #pragma unroll
        for (int n = 0; n < DSN; ++n) {
            hst[n] = __expf(dlt * Av[n]) * hst[n] + dx * sh[n];
            ys += hst[n] * sh[DSN + n];
        }
        ys += xv * Dv;
        int lz = dir ? (L_TOK - 1 - l) : l;       // z read in scan order
        float zv = zP[(size_t)lz * (2 * DI) + c];
        ys *= zv / (1.f + __expf(-zv));           // * silu(z)
        yP[(size_t)lz * DI + c] = ys;             // store un-reversed
        __syncthreads();
    }
}

// ---------------- final residual + RMSNorm + transpose to (e,b,C,HW) ------
__global__ __launch_bounds__(256)
void final_feat_k(const float* __restrict__ h, const float* __restrict__ res,
                  const float* __restrict__ nfw, float* __restrict__ feat)
{
    const int NT = 2 * NB * L_TOK;
    int wv = (blockIdx.x * (blockDim.x >> 5)) + (threadIdx.x >> 5);
    if (wv >= NT) return;
    int e = wv / (NB * L_TOK);
    int bl = wv % (NB * L_TOK);
    int b = bl / L_TOK, l = bl % L_TOK;
    int lane = threadIdx.x & 31;
    size_t base = (size_t)wv * DM;
    float r[6]; float ss = 0.f;
    #pragma unroll
    for (int j = 0; j < 6; ++j) {
        int d = lane + 32 * j;
        float v = h[base + d] + res[base + d];
        r[j] = v; ss += v * v;
    }
    for (int o = 16; o > 0; o >>= 1) ss += __shfl_xor(ss, o, 32);
    float inv = rsqrtf(ss / (float)DM + 1e-5f);
    if (l > 0) {
        int p = l - 1;
        #pragma unroll
        for (int j = 0; j < 6; ++j) {
            int d = lane + 32 * j;
            feat[(((size_t)e * NB + b) * DM + d) * NPATCH + p] = r[j] * inv * nfw[e * DM + d];
        }
    }
}

// ---------------- depthwise 3x3 SAME conv ---------------------------------
__global__ void dwconv_k(const float* __restrict__ in, const float* __restrict__ w,
                         float* __restrict__ out)
{
    int idx = blockIdx.x * blockDim.x + threadIdx.x;
    const int TOT = NB * DM * NPATCH;
    if (idx >= TOT) return;
    int p = idx % NPATCH; int c = (idx / NPATCH) % DM; int b = idx / (NPATCH * DM);
    int py = p / 14, px = p % 14;
    const float* ip = in + ((size_t)b * DM + c) * NPATCH;
    const float* wp = w + c * 9;
    float acc = 0.f;
    #pragma unroll
    for (int dy = -1; dy <= 1; ++dy)
        #pragma unroll
        for (int dx = -1; dx <= 1; ++dx) {
            int yy = py + dy, xx = px + dx;
            if (yy >= 0 && yy < 14 && xx >= 0 && xx < 14)
                acc += ip[yy * 14 + xx] * wp[(dy + 1) * 3 + (dx + 1)];
        }
    out[idx] = acc;
}

// ---------------- L2 normalize each channel row over n=196 ----------------
__global__ __launch_bounds__(256)
void l2norm_k(float* __restrict__ q)
{
    int wv = (blockIdx.x * (blockDim.x >> 5)) + (threadIdx.x >> 5);
    if (wv >= NB * DM) return;
    float* p = q + (size_t)wv * NPATCH;
    int lane = threadIdx.x & 31;
    float ss = 0.f;
    for (int d = lane; d < NPATCH; d += 32) ss += p[d] * p[d];
    for (int o = 16; o > 0; o >>= 1) ss += __shfl_xor(ss, o, 32);
    float inv = 1.f / fmaxf(sqrtf(ss), 1e-12f);
    for (int d = lane; d < NPATCH; d += 32) p[d] *= inv;
}

// ---------------- softmax over 64 with per-head temperature ----------------
__global__ __launch_bounds__(256)
void softmax64_k(float* __restrict__ S, const float* __restrict__ temp)
{
    int wv = (blockIdx.x * (blockDim.x >> 5)) + (threadIdx.x >> 5);
    if (wv >= NB * 3 * 64) return;
    int hd = (wv / 64) % 3;
    float tt = temp[hd];
    float* p = S + (size_t)wv * 64;
    int lane = threadIdx.x & 31;
    float v0 = p[lane] * tt, v1 = p[lane + 32] * tt;
    float mx = fmaxf(v0, v1);
    for (int o = 16; o > 0; o >>= 1) mx = fmaxf(mx, __shfl_xor(mx, o, 32));
    float e0 = __expf(v0 - mx), e1 = __expf(v1 - mx);
    float sm = e0 + e1;
    for (int o = 16; o > 0; o >>= 1) sm += __shfl_xor(sm, o, 32);
    p[lane] = e0 / sm; p[lane + 32] = e1 / sm;
}

// =====================================================================
// host side
// =====================================================================
static void launch_gemm(hipStream_t s,
                        const float* A, int sA, int lda,
                        const float* A2, int sA2,
                        const float* B, int sB, int ldb, int transB,
                        const float* bias, int sBias,
                        const float* Cadd, int sCadd, int ldAdd,
                        float* C, int sC, int ldc,
                        int M, int N, int K, float alpha, int mode, int nz)
{
    int tiles = ((N + 15) / 16) * ((M + 15) / 16);
    dim3 grid((tiles + 3) / 4, 1, nz);
    if (A2)
        gemm16_f32<0,1><<<grid, 128, 0, s>>>(A, sA, lda, A2, sA2, B, sB, ldb,
                                             bias, sBias, Cadd, sCadd, ldAdd,
                                             C, sC, ldc, M, N, K, alpha, mode);
    else if (transB)
        gemm16_f32<1,0><<<grid, 128, 0, s>>>(A, sA, lda, nullptr, 0, B, sB, ldb,
                                             bias, sBias, Cadd, sCadd, ldAdd,
                                             C, sC, ldc, M, N, K, alpha, mode);
    else
        gemm16_f32<0,0><<<grid, 128, 0, s>>>(A, sA, lda, nullptr, 0, B, sB, ldb,
                                             bias, sBias, Cadd, sCadd, ldAdd,
                                             C, sC, ldc, M, N, K, alpha, mode);
}

extern "C" void kernel_launch(void* const* d_in, const int* in_sizes, int n_in,
                              void* d_out, int out_size, void* d_ws, size_t ws_size,
                              hipStream_t stream)
{
    const float* x1      = (const float*)d_in[0];
    const float* x2      = (const float*)d_in[1];
    const float* patch_w = (const float*)d_in[2];
    const float* patch_b = (const float*)d_in[3];
    const float* cls_tok = (const float*)d_in[4];
    const float* pos_emb = (const float*)d_in[5];
    const float* norm_w  = (const float*)d_in[6];
    const float* in_w    = (const float*)d_in[7];
    const float* conv_w  = (const float*)d_in[8];
    const float* conv_b  = (const float*)d_in[9];
    const float* xp_w    = (const float*)d_in[10];
    const float* dt_w    = (const float*)d_in[11];
    const float* dt_b    = (const float*)d_in[12];
    const float* A_log   = (const float*)d_in[13];
    const float* Dpw     = (const float*)d_in[14];
    const float* out_w   = (const float*)d_in[15];
    const float* normf_w = (const float*)d_in[16];
    const float* qw      = (const float*)d_in[17];
    const float* qdw     = (const float*)d_in[18];
    const float* kw      = (const float*)d_in[19];
    const float* kdw     = (const float*)d_in[20];
    const float* vw      = (const float*)d_in[21];
    const float* vdw     = (const float*)d_in[22];
    const float* pww     = (const float*)d_in[23];
    const float* temp    = (const float*)d_in[24];
    float* out = (float*)d_out;
    float* ws  = (float*)d_ws;

    // ---- workspace layout (floats) ----
    const int ROWS = 2 * NB * L_TOK;              // 1576 tokens (both encoders)
    const int MROW = NB * L_TOK;                  // 788 rows per encoder
    size_t o = 0;
    const size_t ROPEC = o; o += (size_t)NPATCH * DM;
    const size_t ROPES = o; o += (size_t)NPATCH * DM;
    const size_t H     = o; o += (size_t)ROWS * DM;
    const size_t RES   = o; o += (size_t)ROWS * DM;
    const size_t HS    = o; o += (size_t)ROWS * DM;
    const size_t XZ    = o; o += (size_t)ROWS * 2 * DI;
    const size_t XC    = o; o += (size_t)2 * ROWS * DI;   // [dir][e][b][l][c]
    const size_t DBL   = o; o += (size_t)2 * ROWS * 48;
    const size_t DELTA = o; o += (size_t)2 * ROWS * DI;
    const size_t YB    = o; o += (size_t)2 * ROWS * DI;
    const size_t FEAT  = o; o += (size_t)2 * NB * DM * NPATCH;
    const size_t TMP   = o; o += (size_t)NB * DM * NPATCH;
    const size_t QB    = o; o += (size_t)NB * DM * NPATCH;
    const size_t KB    = o; o += (size_t)NB * DM * NPATCH;
    const size_t VB    = o; o += (size_t)NB * DM * NPATCH;
    const size_t SB    = o; o += (size_t)NB * 3 * 64 * 64;
    const size_t OB    = o; o += (size_t)NB * DM * NPATCH;
    const size_t IMCOL = DELTA;   // alias: im2col used only before layer loop
    (void)ws_size; (void)in_sizes; (void)n_in; (void)out_size;

    const int DIRSTR = ROWS * DI;                 // dir stride in XC/DELTA/YB
    const int DIRSTR48 = ROWS * 48;

    // ---- prologue ----
    rope_tab_k<<<(NPATCH * DM + 255) / 256, 256, 0, stream>>>(ws + ROPEC, ws + ROPES);
    im2col_k<<<(2 * NB * NPATCH * 768 + 255) / 256, 256, 0, stream>>>(x1, x2, ws + IMCOL);
    init_h_k<<<(ROWS * DM + 255) / 256, 256, 0, stream>>>(cls_tok, pos_emb, ws + H, ws + RES);
    for (int e = 0; e < 2; ++e) {
        // tokens = im2col @ patch_w.T + patch_b + pos_embed[1:], into h rows 1..196
        launch_gemm(stream,
                    ws + IMCOL + (size_t)e * NB * NPATCH * 768, NPATCH * 768, 768,
                    nullptr, 0,
                    patch_w + (size_t)e * DM * 768, 0, 768, 0,
                    patch_b + (size_t)e * DM, 0,
                    pos_emb + (size_t)e * L_TOK * DM + DM, 0, DM,
                    ws + H + (size_t)e * MROW * DM + DM, L_TOK * DM, DM,
                    NPATCH, DM, 768, 1.f, 0, NB);
    }

    // ---- 24 mamba layers (both encoders batched via gridDim.z) ----
    for (int l = 0; l < NDEPTH; ++l) {
        rope_norm_k<<<(ROWS + 7) / 8, 256, 0, stream>>>(
            ws + H, ws + RES, ws + HS, ws + ROPEC, ws + ROPES, norm_w, l);

        // in-proj: xz = hs @ in_w.T      (788x192 x 192x768)
        launch_gemm(stream, ws + HS, MROW * DM, DM, nullptr, 0,
                    in_w + (size_t)l * 2 * DI * DM, NDEPTH * 2 * DI * DM, DM, 0,
                    nullptr, 0, nullptr, 0, 0,
                    ws + XZ, MROW * 2 * DI, 2 * DI,
                    MROW, 2 * DI, DM, 1.f, 0, 2);

        conv_silu_k<<<(2 * ROWS * DI + 255) / 256, 256, 0, stream>>>(
            ws + XZ, conv_w, conv_b, ws + XC, l);

        for (int dir = 0; dir < 2; ++dir) {
            // x-proj: dbl = xc @ xp_w.T  (788x384 x 384x44) -> ldc 48
            launch_gemm(stream, ws + XC + (size_t)dir * DIRSTR, MROW * DI, DI,
                        nullptr, 0,
                        xp_w + ((size_t)l * 2 + dir) * 44 * DI,
                        NDEPTH * 2 * 44 * DI, DI, 0,
                        nullptr, 0, nullptr, 0, 0,
                        ws + DBL + (size_t)dir * DIRSTR48, MROW * 48, 48,
                        MROW, 44, DI, 1.f, 0, 2);
            // dt-proj + softplus: delta = softplus(dbl[:, :12] @ dt_w.T + dt_b)
            launch_gemm(stream, ws + DBL + (size_t)dir * DIRSTR48, MROW * 48, 48,
                        nullptr, 0,
                        dt_w + ((size_t)l * 2 + dir) * DI * DTRK,
                        NDEPTH * 2 * DI * DTRK, DTRK, 0,
                        dt_b + ((size_t)l * 2 + dir) * DI, NDEPTH * 2 * DI,
                        nullptr, 0, 0,
                        ws + DELTA + (size_t)dir * DIRSTR, MROW * DI, DI,
                        MROW, DI, DTRK, 1.f, 1, 2);
        }

        scan_k<<<16, 384, 0, stream>>>(ws + XC, ws + DBL, ws + DELTA, ws + XZ,
                                       A_log, Dpw, ws + YB, l);

        // out-proj: h = 0.5 * (yf + yb) @ out_w.T
        launch_gemm(stream, ws + YB, MROW * DI, DI,
                    ws + YB + (size_t)DIRSTR, MROW * DI,
                    out_w + (size_t)l * DM * DI, NDEPTH * DM * DI, DI, 0,
                    nullptr, 0, nullptr, 0, 0,
                    ws + H, MROW * DM, DM,
                    MROW, DM, DI, 0.5f, 0, 2);
    }

    // ---- final norm + feature transpose ----
    final_feat_k<<<(ROWS + 7) / 8, 256, 0, stream>>>(ws + H, ws + RES, normf_w, ws + FEAT);

    const float* f1 = ws + FEAT;
    const float* f2 = ws + FEAT + (size_t)NB * DM * NPATCH;
    const int FB = DM * NPATCH;   // per-batch feature block

    // ---- cross attention ----
    // q = dwconv(qw @ f1)
    launch_gemm(stream, qw, 0, DM, nullptr, 0, f1, FB, NPATCH, 1,
                nullptr, 0, nullptr, 0, 0, ws + TMP, FB, NPATCH,
                DM, NPATCH, DM, 1.f, 0, NB);
    dwconv_k<<<(NB * FB + 255) / 256, 256, 0, stream>>>(ws + TMP, qdw, ws + QB);
    // k = dwconv(kw @ f2)
    launch_gemm(stream, kw, 0, DM, nullptr, 0, f2, FB, NPATCH, 1,
                nullptr, 0, nullptr, 0, 0, ws + TMP, FB, NPATCH,
                DM, NPATCH, DM, 1.f, 0, NB);
    dwconv_k<<<(NB * FB + 255) / 256, 256, 0, stream>>>(ws + TMP, kdw, ws + KB);
    // v = dwconv(vw @ f2)
    launch_gemm(stream, vw, 0, DM, nullptr, 0, f2, FB, NPATCH, 1,
                nullptr, 0, nullptr, 0, 0, ws + TMP, FB, NPATCH,
                DM, NPATCH, DM, 1.f, 0, NB);
    dwconv_k<<<(NB * FB + 255) / 256, 256, 0, stream>>>(ws + TMP, vdw, ws + VB);

    l2norm_k<<<(NB * DM + 7) / 8, 256, 0, stream>>>(ws + QB);
    l2norm_k<<<(NB * DM + 7) / 8, 256, 0, stream>>>(ws + KB);

    // S = qh @ kh^T  (64x196 x 196x64) per (b,head)
    launch_gemm(stream, ws + QB, 64 * NPATCH, NPATCH, nullptr, 0,
                ws + KB, 64 * NPATCH, NPATCH, 0,
                nullptr, 0, nullptr, 0, 0,
                ws + SB, 64 * 64, 64,
                64, 64, NPATCH, 1.f, 0, NB * 3);
    softmax64_k<<<(NB * 3 * 64 + 7) / 8, 256, 0, stream>>>(ws + SB, temp);
    // out = attn @ vh  (64x64 x 64x196)
    launch_gemm(stream, ws + SB, 64 * 64, 64, nullptr, 0,
                ws + VB, 64 * NPATCH, NPATCH, 1,
                nullptr, 0, nullptr, 0, 0,
                ws + OB, 64 * NPATCH, NPATCH,
                64, NPATCH, 64, 1.f, 0, NB * 3);
    // final projection: d_out = pw @ out   (192x192 x 192x196) per batch
    launch_gemm(stream, pww, 0, DM, nullptr, 0,
                ws + OB, FB, NPATCH, 1,
                nullptr, 0, nullptr, 0, 0,
                out, FB, NPATCH,
                DM, NPATCH, DM, 1.f, 0, NB);
}